// CTCLayer_81664508166369
// MI455X (gfx1250) — compile-verified
//
#include <hip/hip_runtime.h>
#include <hip/hip_bf16.h>
#include <stdint.h>

// CTC forward algorithm (keras ctc_batch_cost semantics).
// B=64, T=2048, C=256 (blank=C-1), L=256, S=2L+1=513.
// One workgroup per batch row; 544 threads = 17 wave32s; alpha lattice in LDS;
// probability rows streamed HBM->LDS via gfx1250 async loads, 16-row ring buffer
// (~16 iterations of HBM latency cover), 2 barriers per lattice step.

#define B_DIM   64
#define T_DIM   2048
#define C_DIM   256
#define L_DIM   256
#define S_DIM   513
#define BLANK_  255
#define NEGINF  (-1e30f)
#define EPS_    1e-7f
#define NBUF    16              // ring depth (rows in flight); 2 async instrs per row
#define NTHREADS 544

// ---- gfx1250 async global->LDS + asynccnt wait ----
#if __has_builtin(__builtin_amdgcn_s_wait_asynccnt)
#define WAIT_ASYNC(n) __builtin_amdgcn_s_wait_asynccnt(n)
#else
#define WAIT_ASYNC(n) asm volatile("s_wait_asynccnt %0" :: "i"(n) : "memory")
#endif

// Builtin parameter types: pointer to 16-byte int vector, AS1 source / AS3 dest.
typedef int v4i __attribute__((vector_size(16)));
typedef __attribute__((address_space(1))) v4i gv4i;
typedef __attribute__((address_space(3))) v4i lv4i;

__device__ __forceinline__ void async_copy_row_1kb(const float* grow, float* ldst, int lane) {
    // 1024 bytes = 256 floats; 32 lanes x 16B x 2 instructions.
    gv4i* gp = (gv4i*)(uintptr_t)(grow + lane * 4);
    lv4i* lp = (lv4i*)(uintptr_t)(ldst + lane * 4);
#if __has_builtin(__builtin_amdgcn_global_load_async_to_lds_b128)
    __builtin_amdgcn_global_load_async_to_lds_b128(gp, lp, 0,   0);
    __builtin_amdgcn_global_load_async_to_lds_b128(gp, lp, 512, 0);
#else
    asm volatile("global_load_async_to_lds_b128 %0, %1, off"
                 :: "v"(lp), "v"(gp) : "memory");
    asm volatile("global_load_async_to_lds_b128 %0, %1, off offset:512"
                 :: "v"(lp), "v"(gp) : "memory");
#endif
}

__device__ __forceinline__ float lse2(float a, float b) {
    float m = fmaxf(a, b);
    float d = fminf(a, b) - m;
    return m + __logf(1.0f + __expf(d));
}

__global__ __launch_bounds__(NTHREADS) void ctc_fwd_kernel(const int* __restrict__ y_true,
                                                           const float* __restrict__ y_pred,
                                                           float* __restrict__ out) {
    __shared__ float s_alphaA[S_DIM + 2];             // index shift +2: pad [0..1] = NEGINF
    __shared__ float s_alphaB[S_DIM + 2];
    __shared__ __align__(16) float s_prob[NBUF][C_DIM];  // raw softmax rows (ring buffer)
    __shared__ int   s_ext[S_DIM];
    __shared__ int   s_len;

    const int tid = threadIdx.x;
    const int b   = blockIdx.x;
    const int*   yt = y_true + b * L_DIM;
    const float* yp = y_pred + (size_t)b * T_DIM * C_DIM;

    if (tid == 0) s_len = 0;
    __syncthreads();

    // ---- prologue: label length, extended sequence, alpha init, prefetch NBUF rows ----
    if (tid < L_DIM) {
        if (yt[tid] >= 0) atomicAdd(&s_len, 1);
    }
    for (int s = tid; s < S_DIM; s += NTHREADS) {
        int e;
        if (s & 1) { int v = yt[(s - 1) >> 1]; e = (v < 0) ? 0 : v; }
        else       { e = BLANK_; }
        s_ext[s] = e;
    }
    for (int i = tid; i < S_DIM + 2; i += NTHREADS) {
        // state 0 lives at index 2; seed log(1)=0 there so the generic step at t=0
        // reproduces the reference alpha0 exactly (skip into s=2 blocked: ext[2]=blank).
        s_alphaA[i] = (i == 2) ? 0.0f : NEGINF;
        s_alphaB[i] = NEGINF;
    }
    if (tid < 32) {
        #pragma unroll
        for (int r = 0; r < NBUF; ++r)
            async_copy_row_1kb(yp + (size_t)r * C_DIM, s_prob[r], tid);
    }
    __syncthreads();

    const int valid_limit = 2 * s_len + 1;

    // per-thread invariants (one lattice state per thread; threads >= S_DIM idle)
    const int  sA   = tid;
    const bool live = (sA < S_DIM);
    const int  eA   = live ? s_ext[sA] : BLANK_;
    const bool skA  = live && (sA >= 2) && (eA != BLANK_) && (eA != s_ext[sA - 2]);
    const bool invA = (sA >= valid_limit);

    float* cur = s_alphaA;
    float* nxt = s_alphaB;

    for (int t = 0; t < T_DIM; ++t) {
        const int buf = t & (NBUF - 1);

        if (tid < 32) {
            // rows t .. t+NBUF-1 may be outstanding (2 instrs each); allowing
            // 2*(NBUF-1) ensures the two oldest (row t's) have completed.
            if (t + NBUF <= T_DIM) WAIT_ASYNC(2 * (NBUF - 1));
            else                   WAIT_ASYNC(0);     // tail: drain everything
        }
        __syncthreads();                              // row t visible; prev alpha visible

        if (live) {
            float a1 = cur[sA + 2];
            float a2 = cur[sA + 1];
            float a3 = skA ? cur[sA] : NEGINF;
            // logaddexp3 fused with emit:  m + log( (e^(a1-m)+e^(a2-m)+e^(a3-m)) * (p+eps) )
            float m  = fmaxf(fmaxf(a1, a2), a3);
            float sm = __expf(a1 - m) + __expf(a2 - m) + __expf(a3 - m);
            float p  = s_prob[buf][eA];
            float nv = m + __logf(sm * (p + EPS_));
            nxt[sA + 2] = invA ? NEGINF : nv;
        }
        __syncthreads();                              // all reads of prob[buf] & writes done

        if (tid < 32 && (t + NBUF) < T_DIM)           // refill the just-freed ring slot
            async_copy_row_1kb(yp + (size_t)(t + NBUF) * C_DIM, s_prob[buf], tid);

        float* tswap = cur; cur = nxt; nxt = tswap;
    }

    __syncthreads();
    if (tid == 0) {
        int len = s_len;
        int i1 = 2 * len;
        int i2 = (2 * len - 1 > 0) ? (2 * len - 1) : 0;
        float f1 = cur[i1 + 2];
        float f2 = cur[i2 + 2];
        out[b] = -lse2(f1, f2);
    }
}

extern "C" void kernel_launch(void* const* d_in, const int* in_sizes, int n_in,
                              void* d_out, int out_size, void* d_ws, size_t ws_size,
                              hipStream_t stream) {
    (void)d_ws; (void)ws_size; (void)n_in; (void)out_size;
    const int*   y_true = (const int*)d_in[0];
    const float* y_pred = (const float*)d_in[1];
    float*       out    = (float*)d_out;
    const int B = in_sizes[0] / L_DIM;                // 64
    ctc_fwd_kernel<<<B, NTHREADS, 0, stream>>>(y_true, y_pred, out);
}